// AggregationLayer_49795850830402
// MI455X (gfx1250) — compile-verified
//
#include <hip/hip_runtime.h>

// CDNA5 / gfx1250: wave32, WMMA 16x16 tiles.
typedef __attribute__((ext_vector_type(2))) float v2f;
typedef __attribute__((ext_vector_type(8))) float v8f;

// ---------------------------------------------------------------------------
// Kernel 1a: seg_start[s] = E for all s in [0, n_seg]
// ---------------------------------------------------------------------------
__global__ void seg_start_init_kernel(int* __restrict__ seg_start, int n_seg, int E) {
    int i = blockIdx.x * blockDim.x + threadIdx.x;
    if (i <= n_seg) seg_start[i] = E;
}

// ---------------------------------------------------------------------------
// Kernel 1b: boundary scan over the sorted segment_ids.
// seg_start[s] = first edge index e with segment_ids[e] >= s.
// ---------------------------------------------------------------------------
__global__ void seg_start_bounds_kernel(const int* __restrict__ seg_ids,
                                        int* __restrict__ seg_start,
                                        int E, int n_seg) {
    int e = blockIdx.x * blockDim.x + threadIdx.x;
    if (e >= E) return;
    int cur  = seg_ids[e];
    int prev = (e == 0) ? -1 : seg_ids[e - 1];
    if (cur > n_seg - 1) cur = n_seg - 1;   // safety clamp
    for (int s = prev + 1; s <= cur; ++s) seg_start[s] = e;
}

// ---------------------------------------------------------------------------
// Kernel 2: one wave per window of 16 consecutive segments.
// Segment-sum as D = A(one-hot 16x4) x B(4x16 features) + C via
// V_WMMA_F32_16X16X4_F32; 4 feature tiles cover D=64.
//
// Layouts (ISA 7.12.2, f32):
//   A 16x4 : lanes 0-15 -> K=0 (.x), K=1 (.y); lanes 16-31 -> K=2 (.x), K=3 (.y).
//            M = lane & 15.
//   B 4x16 : same K split across lane halves; N = lane & 15.
//   C 16x16: vgpr v -> M = v (lanes 0-15) / v+8 (lanes 16-31), N = lane & 15.
//
// Edge/segment metadata is wave-uniform: readfirstlane forces it to SGPRs so
// sid/gidx chunk loads become scalar loads and the loop has scalar branches.
// ---------------------------------------------------------------------------
__global__ void seg_mean_wmma_kernel(const float* __restrict__ LV,        // [N_SRC, 64]
                                     const int*   __restrict__ gidx,      // [E]
                                     const int*   __restrict__ sid,       // [E] sorted
                                     const int*   __restrict__ seg_start, // [n_seg+1]
                                     float*       __restrict__ out,       // [n_seg, 64]
                                     int n_seg, int n_win) {
    const int flat = blockIdx.x * blockDim.x + threadIdx.x;
    const int wid  = __builtin_amdgcn_readfirstlane(flat >> 5);   // wave-uniform SGPR
    const int lane = threadIdx.x & 31;
    if (wid >= n_win) return;        // whole-wave exit: EXEC stays all-1s for WMMA

    const int s_base = wid * 16;
    const int s_top  = (s_base + 16 < n_seg) ? (s_base + 16) : n_seg;
    const int e_begin = __builtin_amdgcn_readfirstlane(seg_start[s_base]);
    const int e_end   = __builtin_amdgcn_readfirstlane(seg_start[s_top]);

    const int  half = lane >> 4;     // 0: lanes 0-15, 1: lanes 16-31
    const int  mn   = lane & 15;     // M for A/C, N for B/C
    const bool hi   = (half != 0);

    v8f acc0 = {}, acc1 = {}, acc2 = {}, acc3 = {};

    const int n_e    = e_end - e_begin;
    const int n_full = n_e & ~3;
    const int e_full = e_begin + n_full;

    // ---- main loop: full 4-edge chunks, no conditionals ----
    for (int e = e_begin; e < e_full; e += 4) {
        // wave-uniform chunk metadata -> scalar loads
        int sr0 = sid[e + 0] - s_base;
        int sr1 = sid[e + 1] - s_base;
        int sr2 = sid[e + 2] - s_base;
        int sr3 = sid[e + 3] - s_base;
        int g0  = gidx[e + 0];
        int g1  = gidx[e + 1];
        int g2  = gidx[e + 2];
        int g3  = gidx[e + 3];

        // per-lane-half selection of K rows (cndmask, no branches)
        int srx = hi ? sr2 : sr0;
        int sry = hi ? sr3 : sr1;
        int rx  = hi ? g2  : g0;
        int ry  = hi ? g3  : g1;

        v2f a;
        a.x = (srx == mn) ? 1.0f : 0.0f;
        a.y = (sry == mn) ? 1.0f : 0.0f;

        const float* r0 = LV + (size_t)rx * 64 + mn;
        const float* r1 = LV + (size_t)ry * 64 + mn;
        v2f b0; b0.x = r0[0];  b0.y = r1[0];
        v2f b1; b1.x = r0[16]; b1.y = r1[16];
        v2f b2; b2.x = r0[32]; b2.y = r1[32];
        v2f b3; b3.x = r0[48]; b3.y = r1[48];

        acc0 = __builtin_amdgcn_wmma_f32_16x16x4_f32(false, a, false, b0, (short)0, acc0, false, false);
        acc1 = __builtin_amdgcn_wmma_f32_16x16x4_f32(false, a, false, b1, (short)0, acc1, false, false);
        acc2 = __builtin_amdgcn_wmma_f32_16x16x4_f32(false, a, false, b2, (short)0, acc2, false, false);
        acc3 = __builtin_amdgcn_wmma_f32_16x16x4_f32(false, a, false, b3, (short)0, acc3, false, false);
    }

    // ---- epilogue: remainder chunk (uniform trip count 1..3), runs once ----
    const int r = n_e & 3;
    if (r) {
        int sr[4] = { -1, -1, -1, -1 };
        int g[4];
        const int g_safe = gidx[e_begin];          // valid row, finite data
        g[0] = g[1] = g[2] = g[3] = g_safe;
        for (int k = 0; k < r; ++k) {              // uniform bound -> scalar branch
            sr[k] = sid[e_full + k] - s_base;
            g[k]  = gidx[e_full + k];
        }

        int srx = hi ? sr[2] : sr[0];
        int sry = hi ? sr[3] : sr[1];
        int rx  = hi ? g[2]  : g[0];
        int ry  = hi ? g[3]  : g[1];

        v2f a;
        a.x = (srx == mn) ? 1.0f : 0.0f;
        a.y = (sry == mn) ? 1.0f : 0.0f;

        const float* r0 = LV + (size_t)rx * 64 + mn;
        const float* r1 = LV + (size_t)ry * 64 + mn;
        v2f b0; b0.x = r0[0];  b0.y = r1[0];
        v2f b1; b1.x = r0[16]; b1.y = r1[16];
        v2f b2; b2.x = r0[32]; b2.y = r1[32];
        v2f b3; b3.x = r0[48]; b3.y = r1[48];

        acc0 = __builtin_amdgcn_wmma_f32_16x16x4_f32(false, a, false, b0, (short)0, acc0, false, false);
        acc1 = __builtin_amdgcn_wmma_f32_16x16x4_f32(false, a, false, b1, (short)0, acc1, false, false);
        acc2 = __builtin_amdgcn_wmma_f32_16x16x4_f32(false, a, false, b2, (short)0, acc2, false, false);
        acc3 = __builtin_amdgcn_wmma_f32_16x16x4_f32(false, a, false, b3, (short)0, acc3, false, false);
    }

    // ---- scale by 1/max(count,1) and store ----
    if (s_top == s_base + 16) {
        // full window (common case): unconditional stores
#pragma unroll
        for (int v = 0; v < 8; ++v) {
            const int s   = s_base + v + half * 8;
            const int cnt = seg_start[s + 1] - seg_start[s];
            const float inv = 1.0f / (float)(cnt > 1 ? cnt : 1);
            float* o = out + (size_t)s * 64 + mn;
            o[0]  = acc0[v] * inv;
            o[16] = acc1[v] * inv;
            o[32] = acc2[v] * inv;
            o[48] = acc3[v] * inv;
        }
    } else {
        // ragged last window
#pragma unroll
        for (int v = 0; v < 8; ++v) {
            const int s = s_base + v + half * 8;
            if (s < n_seg) {
                const int cnt = seg_start[s + 1] - seg_start[s];
                const float inv = 1.0f / (float)(cnt > 1 ? cnt : 1);
                float* o = out + (size_t)s * 64 + mn;
                o[0]  = acc0[v] * inv;
                o[16] = acc1[v] * inv;
                o[32] = acc2[v] * inv;
                o[48] = acc3[v] * inv;
            }
        }
    }
}

// ---------------------------------------------------------------------------
// Host launcher
// ---------------------------------------------------------------------------
extern "C" void kernel_launch(void* const* d_in, const int* in_sizes, int n_in,
                              void* d_out, int out_size, void* d_ws, size_t ws_size,
                              hipStream_t stream) {
    const float* LV   = (const float*)d_in[0];   // layer_values [N_SRC*64]
    const int*   gidx = (const int*)d_in[1];     // gather_idx   [E]
    const int*   sid  = (const int*)d_in[2];     // segment_ids  [E], sorted
    float*       out  = (float*)d_out;

    const int E     = in_sizes[1];
    const int n_seg = out_size / 64;             // output is [n_seg, 64]

    int* seg_start = (int*)d_ws;                 // (n_seg + 1) ints of scratch

    // 1a) init starts to E
    {
        int t = n_seg + 1;
        seg_start_init_kernel<<<(t + 255) / 256, 256, 0, stream>>>(seg_start, n_seg, E);
    }
    // 1b) boundary scan
    seg_start_bounds_kernel<<<(E + 255) / 256, 256, 0, stream>>>(sid, seg_start, E, n_seg);

    // 2) WMMA segment-mean: one wave per 16-segment window
    {
        int n_win = (n_seg + 15) / 16;
        long long threads = (long long)n_win * 32;
        int blocks = (int)((threads + 255) / 256);
        seg_mean_wmma_kernel<<<blocks, 256, 0, stream>>>(LV, gidx, sid, seg_start,
                                                         out, n_seg, n_win);
    }
}